// RelationAggregator_63582695850894
// MI455X (gfx1250) — compile-verified
//
#include <hip/hip_runtime.h>

typedef float v2f __attribute__((ext_vector_type(2)));
typedef float v8f __attribute__((ext_vector_type(8)));

#define N_NODES 100000
#define E_EDGES 600000
#define R_REL   3
#define DIM     128        // D_IN == D_OUT == 128

// -------------------------------------------------------------------------
// out[n, f] = b[f]   (bias broadcast init; gemm_acc then accumulates on top)
// -------------------------------------------------------------------------
__global__ void rgcn_init_bias(float* __restrict__ out,
                               const float* __restrict__ b) {
    size_t idx = (size_t)blockIdx.x * blockDim.x + threadIdx.x;
    if (idx < (size_t)N_NODES * DIM) {
        out[idx] = b[idx & (DIM - 1)];
    }
}

// -------------------------------------------------------------------------
// Edge scatter for one relation:
//   agg[dst[e], :] += vals[e] * x[src[e], :]
// One wave32 per edge; each lane owns 4 contiguous features (float4 gather,
// 4x global_atomic_add_f32 scatter). 8 edges per 256-thread block.
// x and agg stay L2-resident (51 MB each, global L2 = 192 MB).
// -------------------------------------------------------------------------
__global__ void rgcn_scatter(const float* __restrict__ x,
                             const int*   __restrict__ src,
                             const int*   __restrict__ dst,
                             const float* __restrict__ vals,
                             float*       __restrict__ agg) {
    const int wave = threadIdx.x >> 5;    // 0..7
    const int lane = threadIdx.x & 31;
    const long long e = (long long)blockIdx.x * 8 + wave;
    if (e >= E_EDGES) return;

    const int   s = src[e];
    const int   d = dst[e];
    const float v = vals[e];

    const float4 m = ((const float4*)(x + (size_t)s * DIM))[lane];
    float* ar = agg + (size_t)d * DIM + lane * 4;
    atomicAdd(ar + 0, v * m.x);
    atomicAdd(ar + 1, v * m.y);
    atomicAdd(ar + 2, v * m.z);
    atomicAdd(ar + 3, v * m.w);
}

// -------------------------------------------------------------------------
// out[N,128] += A[N,128] @ W[128,128]^T   via V_WMMA_F32_16X16X4_F32.
//
// Block = 256 threads = 8 waves. Wave w computes the 16x16 C tile at
// (row0 = 16*blockIdx.x, n0 = 16*w); K-loop runs 128/4 = 32 WMMAs.
//
// Fragment layouts (ISA 7.12.2, wave32):
//   A 16x4 fp32 : lane<16 -> {A[M=lane,k], A[M=lane,k+1]}
//                 lane>=16 -> {A[M=lane-16,k+2], A[M=lane-16,k+3]}
//                 => contiguous float2 at A[(row0+lo)*128 + k + 2*hi]
//   B 4x16 fp32 : symmetric; B[kk,n] = W[n,kk]
//                 => contiguous float2 at W[(n0+lo)*128 + k + 2*hi]
//   C/D 16x16   : VGPR i <-> M = i + 8*hi, N = lo
// -------------------------------------------------------------------------
__global__ void rgcn_gemm_acc(const float* __restrict__ A,
                              const float* __restrict__ W,
                              float*       __restrict__ out) {
    const int lane = threadIdx.x & 31;
    const int wave = threadIdx.x >> 5;
    const int hi   = lane >> 4;          // 0 or 1
    const int lo   = lane & 15;
    const int row0 = blockIdx.x * 16;
    const int n0   = wave * 16;

    // Load accumulator tile from out (accumulate semantics).
    v8f c;
#pragma unroll
    for (int i = 0; i < 8; ++i) {
        const int m = i + 8 * hi;
        c[i] = out[(size_t)(row0 + m) * DIM + n0 + lo];
    }

    const float* arow = A + (size_t)(row0 + lo) * DIM + 2 * hi;
    const float* wrow = W + (size_t)(n0  + lo) * DIM + 2 * hi;

#pragma unroll
    for (int k = 0; k < DIM; k += 4) {
        v2f a = *(const v2f*)(arow + k);
        v2f b = *(const v2f*)(wrow + k);
        // 8 args: (neg_a, A, neg_b, B, c_mod, C, reuse_a, reuse_b)
        c = __builtin_amdgcn_wmma_f32_16x16x4_f32(
                false, a, false, b, (short)0, c, false, false);
    }

#pragma unroll
    for (int i = 0; i < 8; ++i) {
        const int m = i + 8 * hi;
        out[(size_t)(row0 + m) * DIM + n0 + lo] = c[i];
    }
}

// -------------------------------------------------------------------------
// Host-side launcher. Inputs (setup_inputs order):
//   0: x      [N,128]   f32
//   1: src    [R,E]     i32
//   2: dst    [R,E]     i32
//   3: vals   [R,E]     f32
//   4: W_self [128,128] f32
//   5: b_self [128]     f32
//   6: W_rels [R,128,128] f32
// d_out: [N,128] f32.  d_ws: one [N,128] f32 agg buffer (51.2 MB).
// -------------------------------------------------------------------------
extern "C" void kernel_launch(void* const* d_in, const int* in_sizes, int n_in,
                              void* d_out, int out_size, void* d_ws, size_t ws_size,
                              hipStream_t stream) {
    const float* x      = (const float*)d_in[0];
    const int*   src    = (const int*)  d_in[1];
    const int*   dst    = (const int*)  d_in[2];
    const float* vals   = (const float*)d_in[3];
    const float* W_self = (const float*)d_in[4];
    const float* b_self = (const float*)d_in[5];
    const float* W_rels = (const float*)d_in[6];
    float* out = (float*)d_out;
    float* agg = (float*)d_ws;           // N*DIM floats

    const size_t aggBytes = (size_t)N_NODES * DIM * sizeof(float);

    // 1) out = b (broadcast)
    {
        const int total = N_NODES * DIM;
        rgcn_init_bias<<<(total + 255) / 256, 256, 0, stream>>>(out, b_self);
    }

    // 2) out += x @ W_self^T
    rgcn_gemm_acc<<<N_NODES / 16, 256, 0, stream>>>(x, W_self, out);

    // 3) per relation: zero agg -> scatter -> out += agg @ W_r^T
    for (int r = 0; r < R_REL; ++r) {
        hipMemsetAsync(agg, 0, aggBytes, stream);
        rgcn_scatter<<<E_EDGES / 8, 256, 0, stream>>>(
            x, src + (size_t)r * E_EDGES, dst + (size_t)r * E_EDGES,
            vals + (size_t)r * E_EDGES, agg);
        rgcn_gemm_acc<<<N_NODES / 16, 256, 0, stream>>>(
            agg, W_rels + (size_t)r * DIM * DIM, out);
    }
}